// DYSPNAccelerate_80960133529727
// MI455X (gfx1250) — compile-verified
//
#include <hip/hip_runtime.h>
#include <stdint.h>

// ---- problem constants (from reference) -----------------------------------
#define BS     4
#define H      256
#define W      320
#define HW     (H * W)
#define KK     49
#define MID    24
#define ITERS  6

// ---- tiling ---------------------------------------------------------------
#define VEC    4                   // pixels per thread (one b128 lane quad)
#define TXT    16                  // threads in x
#define TYT    8                   // threads in y  -> 128 threads = 4 wave32
#define TPX    (TXT * VEC)         // 64 pixels per tile in x
#define TPY    TYT                 // 8 rows per tile
#define HALO   3
#define LDS_WU (TPX + 2 * HALO)    // 70 used columns
#define LDS_W  72                  // padded row stride (floats)
#define LDS_H  (TPY + 2 * HALO)    // 14 rows

typedef float v4f __attribute__((ext_vector_type(4)));

// ring id per 7x7 tap (concentric squares), matches reference _RING
__device__ __constant__ const int RING[KK] = {
    3,3,3,3,3,3,3,
    3,2,2,2,2,2,3,
    3,2,1,1,1,2,3,
    3,2,1,0,1,2,3,
    3,2,1,1,1,2,3,
    3,2,2,2,2,2,3,
    3,3,3,3,3,3,3};

// gfx1250 async global->LDS copy, 4 bytes per enabled lane (ASYNCcnt).
__device__ __forceinline__ void async_g2lds_b32(uint32_t lds_byte_off,
                                                const float* gptr) {
    uint64_t ga = (uint64_t)(uintptr_t)gptr;
    asm volatile("global_load_async_to_lds_b32 %0, %1, off"
                 :
                 : "v"(lds_byte_off), "v"(ga)
                 : "memory");
}

__device__ __forceinline__ void wait_asynccnt0() {
    asm volatile("s_wait_asynccnt 0" ::: "memory");
}

__global__ __launch_bounds__(TXT * TYT)
void dyspn_kernel(const float* __restrict__ kern,    // [BS][KK][HW]
                  const float* __restrict__ input,   // [BS][HW]
                  const float* __restrict__ input0,  // [BS][HW]
                  const float* __restrict__ att,     // [BS][ITERS][4][HW]
                  const int*   __restrict__ iter_idx,
                  float*       __restrict__ out) {   // [BS][HW]
    __shared__ float s_tile[LDS_H * LDS_W];

    const int tx  = threadIdx.x;             // 0..15
    const int ty  = threadIdx.y;             // 0..7
    const int tid = ty * TXT + tx;           // 0..127
    const int bx0 = blockIdx.x * TPX;        // tile origin x (pixels)
    const int by0 = blockIdx.y * TPY;        // tile origin y
    const int b   = blockIdx.z;

    // ---- stage input tile (+halo) into LDS with async copies --------------
    // element (ly,lx) = input[b][by0-3+ly][bx0-3+lx], zero outside the image.
    for (int e = tid; e < LDS_H * LDS_WU; e += TXT * TYT) {
        const int ly = e / LDS_WU;
        const int lx = e - ly * LDS_WU;
        const int gy = by0 - HALO + ly;
        const int gx = bx0 - HALO + lx;
        const uint32_t loff = (uint32_t)(ly * LDS_W + lx) * 4u;
        if (gy >= 0 && gy < H && gx >= 0 && gx < W) {
            async_g2lds_b32(loff, input + (size_t)b * HW + gy * W + gx);
        } else {
            s_tile[ly * LDS_W + lx] = 0.0f;   // zero pad (DS store)
        }
    }
    wait_asynccnt0();      // this wave's async LDS writes have landed
    __syncthreads();       // compiler also waits its own DScnt here

    // ---- per-quad (4 pixels) reduction ------------------------------------
    const int px = bx0 + tx * VEC;           // first pixel of the quad
    const int py = by0 + ty;
    const int p  = py * W + px;              // multiple of 4 -> 16B aligned

    const int it = *iter_idx;                // uniform scalar (i == 2)

    v4f a[4];
#pragma unroll
    for (int r = 0; r < 4; ++r)
        a[r] = *(const v4f*)(att + (size_t)(((b * ITERS + it) * 4) + r) * HW + p);

    const float* kb     = kern + (size_t)b * KK * HW + p;
    const v4f    center = *(const v4f*)(input0 + (size_t)b * HW + p);

    v4f acc = (v4f)0.0f;
    v4f nrm = (v4f)0.0f;

#pragma unroll
    for (int dy = 0; dy < 7; ++dy) {
        // 10 consecutive tile floats cover all 4 pixels x 7 taps of this row.
        float row[VEC + 6];
#pragma unroll
        for (int c = 0; c < VEC + 6; ++c)
            row[c] = s_tile[(ty + dy) * LDS_W + tx * VEC + c];

#pragma unroll
        for (int dx = 0; dx < 7; ++dx) {
            const int j = dy * 7 + dx;
            // kernel tensor streamed exactly once: 16B/lane non-temporal load
            const v4f kj = __builtin_nontemporal_load((const v4f*)(kb + (size_t)j * HW));
            const v4f wj = kj * a[RING[j]];
#pragma unroll
            for (int q = 0; q < VEC; ++q) {
                const float pj = (j == MID) ? center[q] : row[dx + q];
                acc[q] = __builtin_fmaf(pj, wj[q], acc[q]);
                nrm[q] += __builtin_fabsf(wj[q]);
            }
        }
    }

    const v4f res = acc / nrm;
    __builtin_nontemporal_store(res, (v4f*)(out + (size_t)b * HW + p));
}

extern "C" void kernel_launch(void* const* d_in, const int* in_sizes, int n_in,
                              void* d_out, int out_size, void* d_ws, size_t ws_size,
                              hipStream_t stream) {
    (void)in_sizes; (void)n_in; (void)out_size; (void)d_ws; (void)ws_size;
    const float* kern   = (const float*)d_in[0];  // (4, 49, 81920)
    const float* input  = (const float*)d_in[1];  // (4, 1, 256, 320)
    const float* input0 = (const float*)d_in[2];  // (4, 1, 256, 320)
    const float* att    = (const float*)d_in[3];  // (4, 6, 4, 81920)
    const int*   it     = (const int*)d_in[4];    // scalar i
    float*       out    = (float*)d_out;          // (4, 1, 256, 320)

    dim3 block(TXT, TYT, 1);                // 128 threads = 4 wave32
    dim3 grid(W / TPX, H / TPY, BS);        // (5, 32, 4) = 640 blocks
    dyspn_kernel<<<grid, block, 0, stream>>>(kern, input, input0, att, it, out);
}